// GridSamplePScan_54855322304932
// MI455X (gfx1250) — compile-verified
//
#include <hip/hip_runtime.h>
#include <hip/hip_bf16.h>
#include <math.h>

// Problem constants from the reference: B=2, L=16, C=16, H=64, W=64.
#define BB 2
#define LL 16
#define CC 16
#define HH 64
#define WW 64
#define HW (HH * WW)      // 4096
#define HW2 (2 * HW)      // 8192 (two flow components)

typedef __attribute__((ext_vector_type(2))) float v2f;
typedef __attribute__((ext_vector_type(8))) float v8f;

// -------------------------------------------------------------------------
// Kernel 1: cum = tril(ones(16,16)) @ flows  via V_WMMA_F32_16X16X4_F32.
// Each wave produces a 16x16 tile of cum: rows = all 16 layers, cols = 16
// consecutive flat indices of the (2,H,W) axis. A (tril ones) is synthesized
// in registers; B comes from flows. 4 k-steps of 16x16x4 accumulate in c.
//
// Layouts (ISA 7.12.2, wave32):
//   A 16x4 f32 : lane l holds row M=l%16; VGPR0 = K (4k + 2*(l>=16)),
//                VGPR1 = K+1.
//   B 4x16 f32 : lane l holds col N=l%16; VGPR0 = row K (4k + 2*(l>=16)),
//                VGPR1 = row K+1.
//   C/D 16x16  : VGPR r holds row r (lanes 0-15) / row r+8 (lanes 16-31),
//                col N = lane%16.
// -------------------------------------------------------------------------
__global__ void cumsum_wmma_kernel(const float* __restrict__ flows,
                                   float* __restrict__ cum) {
    int wv   = (blockIdx.x * blockDim.x + threadIdx.x) >> 5; // 0..1023
    int lane = threadIdx.x & 31;
    int half = lane >> 4;   // 0 or 1
    int n    = lane & 15;   // row of A held by this lane; also col N of B/D
    int b    = wv >> 9;     // 512 column tiles per batch
    int tile = wv & 511;
    int col  = tile * 16 + n;

    const float* fb = flows + (size_t)b * LL * HW2;

    v8f c = {};
#pragma unroll
    for (int k = 0; k < 4; ++k) {
        int K0 = 4 * k + 2 * half;
        v2f a, bv;
        // tril(ones): M[row][colK] = (colK <= row)
        a.x = (K0     <= n) ? 1.0f : 0.0f;
        a.y = (K0 + 1 <= n) ? 1.0f : 0.0f;
        bv.x = fb[(size_t)K0       * HW2 + col];
        bv.y = fb[(size_t)(K0 + 1) * HW2 + col];
        // (neg_a, A, neg_b, B, c_mod, C, reuse_a, reuse_b)
        c = __builtin_amdgcn_wmma_f32_16x16x4_f32(
                false, a, false, bv, (short)0, c, false, false);
    }

    float* cb = cum + (size_t)b * LL * HW2;
#pragma unroll
    for (int r = 0; r < 8; ++r) {
        int row = r + 8 * half;
        cb[(size_t)row * HW2 + col] = c[r];
    }
}

// -------------------------------------------------------------------------
// Kernel 2: for each (b,i,pixel): sum over j<=i of the bilinear sample of
// images[b,j,:] at grid (gx_base + relx, gy_base + rely), relx/rely from cum,
// gx wrapped into [-1,1), out-of-range taps dropped (weight forced to 0).
// One thread = one pixel; 16-channel f32 accumulator in registers.
// -------------------------------------------------------------------------
__global__ void warp_accum_kernel(const float* __restrict__ images,
                                  const float* __restrict__ cum,
                                  float* __restrict__ out) {
    int chunk = blockIdx.x & 15;          // 16 pixel-chunks of 256
    int i     = (blockIdx.x >> 4) & 15;   // layer i
    int b     = blockIdx.x >> 8;          // batch
    int p     = chunk * 256 + threadIdx.x;
    int h     = p >> 6;
    int w     = p & 63;

    const float* cb = cum + (size_t)b * LL * HW2;
    float cxi = cb[(size_t)(i * 2 + 0) * HW + p];
    float cyi = cb[(size_t)(i * 2 + 1) * HW + p];

    float acc[CC];
#pragma unroll
    for (int c = 0; c < CC; ++c) acc[c] = 0.0f;

    float gxb = ((float)w + 0.5f) * (2.0f / WW) - 1.0f;
    float gyb = ((float)h + 0.5f) * (2.0f / HH) - 1.0f;

    for (int j = 0; j <= i; ++j) {
        float relx = cxi - cb[(size_t)(j * 2 + 0) * HW + p];
        float rely = cyi - cb[(size_t)(j * 2 + 1) * HW + p];
        float gx = gxb + relx;
        float gy = gyb + rely;
        // gx = remainder(gx + 1, 2) - 1  (floor-mod, result in [-1,1))
        float t = gx + 1.0f;
        t  = t - 2.0f * floorf(t * 0.5f);
        gx = t - 1.0f;

        float ix = ((gx + 1.0f) * (float)WW - 1.0f) * 0.5f;
        float iy = ((gy + 1.0f) * (float)HH - 1.0f) * 0.5f;
        float x0f = floorf(ix), y0f = floorf(iy);
        float wx = ix - x0f,    wy = iy - y0f;
        int   x0 = (int)x0f,    y0 = (int)y0f;

        // Fold per-tap validity into the weights (invalid tap contributes 0).
        float wx0 = (x0 >= 0     && x0 < WW)     ? (1.0f - wx) : 0.0f;
        float wx1 = (x0 + 1 >= 0 && x0 + 1 < WW) ? wx          : 0.0f;
        float wy0 = (y0 >= 0     && y0 < HH)     ? (1.0f - wy) : 0.0f;
        float wy1 = (y0 + 1 >= 0 && y0 + 1 < HH) ? wy          : 0.0f;

        int xc0 = min(max(x0, 0), WW - 1),     xc1 = min(max(x0 + 1, 0), WW - 1);
        int yc0 = min(max(y0, 0), HH - 1),     yc1 = min(max(y0 + 1, 0), HH - 1);
        int l00 = yc0 * WW + xc0, l01 = yc0 * WW + xc1;
        int l10 = yc1 * WW + xc0, l11 = yc1 * WW + xc1;

        float w00 = wx0 * wy0, w01 = wx1 * wy0;
        float w10 = wx0 * wy1, w11 = wx1 * wy1;

        const float* img = images + ((size_t)(b * LL + j) * CC) * HW;
#pragma unroll
        for (int c = 0; c < CC; ++c) {
            const float* ic = img + (size_t)c * HW;
            acc[c] += w00 * ic[l00] + w01 * ic[l01]
                    + w10 * ic[l10] + w11 * ic[l11];
        }
    }

    float* ob = out + ((size_t)(b * LL + i) * CC) * HW;
#pragma unroll
    for (int c = 0; c < CC; ++c) ob[(size_t)c * HW + p] = acc[c];
}

extern "C" void kernel_launch(void* const* d_in, const int* in_sizes, int n_in,
                              void* d_out, int out_size, void* d_ws, size_t ws_size,
                              hipStream_t stream) {
    const float* flows  = (const float*)d_in[0];  // (B, L, 2, H, W) f32
    const float* images = (const float*)d_in[1];  // (B, L, C, H, W) f32
    float*       out    = (float*)d_out;          // (B, L, C, H, W) f32
    float*       cumbuf = (float*)d_ws;           // (B, L, 2, H, W) f32 = 1 MB

    // 1024 waves = B * 512 column tiles, each a 16x16 WMMA-tiled tril-matmul.
    cumsum_wmma_kernel<<<128, 256, 0, stream>>>(flows, cumbuf);
    // B * L * (HW/256) = 512 blocks; one thread per pixel of one (b, i).
    warp_accum_kernel<<<512, 256, 0, stream>>>(images, cumbuf, out);
}